// MultiBoxLoss_45775761441078
// MI455X (gfx1250) — compile-verified
//
#include <hip/hip_runtime.h>
#include <math.h>

// ---- problem constants (from reference setup) ----
#define NB 32          // batch
#define NP 8732        // priors
#define NC 81          // classes
#define NO 16          // objects per image
#define THRESH 0.5f
#define SORT_SZ 16384  // next pow2 >= NP

#define TPB 256
#define PPB 128                      // priors per block in k_main
#define TILE_DW (PPB * NC)           // 10368 dwords = 41472 B LDS tile

typedef __attribute__((ext_vector_type(2))) float v2f;
typedef __attribute__((ext_vector_type(8))) float v8f;

__device__ __forceinline__ float iou_xy(float ax0, float ay0, float ax1, float ay1,
                                        float bx0, float by0, float bx1, float by1) {
    float lx = fmaxf(ax0, bx0), ly = fmaxf(ay0, by0);
    float rx = fminf(ax1, bx1), ry = fminf(ay1, by1);
    float w = fmaxf(rx - lx, 0.0f), h = fmaxf(ry - ly, 0.0f);
    float inter = w * h;
    float aa = (ax1 - ax0) * (ay1 - ay0);
    float ab = (bx1 - bx0) * (by1 - by0);
    return inter / (aa + ab - inter);
}

// ---------------- kernel 0: zero accumulators ----------------
__global__ void k_init(float* npos_row, float* accs) {
    int t = threadIdx.x;
    if (t < NB) npos_row[t] = 0.0f;
    if (t < 4) accs[t] = 0.0f;
}

// ---------------- kernel 1: per-prior max/argmax IoU over objects ----------------
__global__ void k_overlap(const float* __restrict__ boxes,        // (N,O,4) xy
                          const float* __restrict__ priors_cxcy,  // (P,4)
                          float* __restrict__ overlap_best,       // (N,P)
                          int* __restrict__ obj_best) {           // (N,P)
    __shared__ float sb[NO * 4];
    int n = blockIdx.y;
    if (threadIdx.x < NO * 4) sb[threadIdx.x] = boxes[n * NO * 4 + threadIdx.x];
    __syncthreads();
    int p = blockIdx.x * blockDim.x + threadIdx.x;
    if (p >= NP) return;
    float cx = priors_cxcy[p * 4 + 0], cy = priors_cxcy[p * 4 + 1];
    float pw = priors_cxcy[p * 4 + 2], ph = priors_cxcy[p * 4 + 3];
    float px0 = cx - 0.5f * pw, py0 = cy - 0.5f * ph;
    float px1 = cx + 0.5f * pw, py1 = cy + 0.5f * ph;
    float best = -1.0f; int bidx = 0;
    for (int o = 0; o < NO; ++o) {
        float v = iou_xy(sb[o * 4 + 0], sb[o * 4 + 1], sb[o * 4 + 2], sb[o * 4 + 3],
                         px0, py0, px1, py1);
        if (v > best) { best = v; bidx = o; }   // first-index tie-break (argmax)
    }
    overlap_best[n * NP + p] = best;
    obj_best[n * NP + p] = bidx;
}

// ---------------- kernel 2: per-object argmax IoU over priors ----------------
__global__ void k_pfo(const float* __restrict__ boxes,
                      const float* __restrict__ priors_cxcy,
                      int* __restrict__ pfo) {                   // (N,O)
    __shared__ float sval[256];
    __shared__ int   sidx[256];
    int o = blockIdx.x, n = blockIdx.y, t = threadIdx.x;
    float bx0 = boxes[(n * NO + o) * 4 + 0], by0 = boxes[(n * NO + o) * 4 + 1];
    float bx1 = boxes[(n * NO + o) * 4 + 2], by1 = boxes[(n * NO + o) * 4 + 3];
    float best = -1.0f; int bidx = NP;
    for (int p = t; p < NP; p += 256) {
        float cx = priors_cxcy[p * 4 + 0], cy = priors_cxcy[p * 4 + 1];
        float pw = priors_cxcy[p * 4 + 2], ph = priors_cxcy[p * 4 + 3];
        float v = iou_xy(bx0, by0, bx1, by1,
                         cx - 0.5f * pw, cy - 0.5f * ph, cx + 0.5f * pw, cy + 0.5f * ph);
        if (v > best) { best = v; bidx = p; }
    }
    sval[t] = best; sidx[t] = bidx;
    __syncthreads();
    for (int s = 128; s > 0; s >>= 1) {
        if (t < s) {
            float ov = sval[t + s]; int oi = sidx[t + s];
            if (ov > sval[t] || (ov == sval[t] && oi < sidx[t])) { sval[t] = ov; sidx[t] = oi; }
        }
        __syncthreads();
    }
    if (t == 0) pfo[n * NO + o] = sidx[0];
}

// ---------------- kernel 3: force-match scatter (sequential per row) ----------------
__global__ void k_assign(const int* __restrict__ pfo,
                         float* __restrict__ overlap_best,
                         int* __restrict__ obj_best) {
    int n = threadIdx.x;
    if (n >= NB) return;
    for (int o = 0; o < NO; ++o) {
        int p = pfo[n * NO + o];
        obj_best[n * NP + p] = o;
        overlap_best[n * NP + p] = 1.0f;
    }
}

__device__ __forceinline__ float vsel8(v8f v, int j) {
    float r = v[0];
    r = (j == 1) ? v[1] : r;
    r = (j == 2) ? v[2] : r;
    r = (j == 3) ? v[3] : r;
    r = (j == 4) ? v[4] : r;
    r = (j == 5) ? v[5] : r;
    r = (j == 6) ? v[6] : r;
    r = (j == 7) ? v[7] : r;
    return r;
}

// ---------------- kernel 4: main streaming kernel ----------------
// Block stages a contiguous 128x81 f32 score tile into LDS with async-tensor copies,
// then each wave32 computes 16 softmax denominators with V_WMMA_F32_16X16X4_F32
// (A = exp(logits) 16x4, B = ones; D rows broadcast per-prior sum(exp)).
__global__ void k_main(const float* __restrict__ plocs,    // (N,P,4)
                       const float* __restrict__ scores,   // (N,P,C)
                       const float* __restrict__ boxes,    // (N,O,4)
                       const int* __restrict__ labels,     // (N,O)
                       const float* __restrict__ priors_cxcy,
                       const float* __restrict__ overlap_best,
                       const int* __restrict__ obj_best,
                       float* __restrict__ ce_neg,         // (N,P)
                       float* __restrict__ npos_row,       // (N)
                       float* __restrict__ accs) {         // [0]=loc, [1]=conf_pos
    __shared__ float stile[TILE_DW];

    const int n = blockIdx.y;
    const int t = threadIdx.x;
    const int pbase = blockIdx.x * PPB;

    // ---- async stage: global -> LDS, coalesced b128 streams (ASYNCcnt) ----
    const unsigned long long gbase =
        (unsigned long long)(uintptr_t)(scores + ((size_t)n * NP + pbase) * NC);
    const unsigned long long gend =
        (unsigned long long)(uintptr_t)(scores + (size_t)NB * NP * NC);
    const unsigned ldsbase = (unsigned)(uintptr_t)stile;  // low 32 bits = LDS offset
#pragma unroll
    for (int it = 0; it < 10; ++it) {
        unsigned off = (unsigned)(it * (TPB * 16) + t * 16);
        unsigned long long ga = gbase + off;
        unsigned long long gc = gend - 16ull;
        ga = (ga > gc) ? gc : ga;
        asm volatile("global_load_async_to_lds_b128 %0, %1, off"
                     :: "v"(ldsbase + off), "v"(ga) : "memory");
    }
    if (t < (TILE_DW - 10 * TPB * 4)) {   // 128-dword tail
        unsigned off = (unsigned)(10 * TPB * 16 + t * 4);
        unsigned long long ga = gbase + off;
        unsigned long long gc = gend - 4ull;
        ga = (ga > gc) ? gc : ga;
        asm volatile("global_load_async_to_lds_b32 %0, %1, off"
                     :: "v"(ldsbase + off), "v"(ga) : "memory");
    }
    asm volatile("s_wait_asynccnt 0x0" ::: "memory");
    __syncthreads();

    // ---- WMMA class-sum: no lane-divergent control flow anywhere here ----
    const int lane = t & 31;
    const int r    = lane & 15;       // A-matrix row M (prior within wave group)
    const int h    = lane >> 4;       // half: K pair {0,1} or {2,3}
    const int lpri = (t >> 5) * 16 + r;      // local prior in tile
    const int p    = pbase + lpri;
    const bool valid = (p < NP);
    const float vmask = valid ? 1.0f : 0.0f;
    const float* lrow = &stile[lpri * NC];

    v8f acc = {};
    v2f bones; bones[0] = 1.0f; bones[1] = 1.0f;
#pragma unroll
    for (int c = 0; c < 20; ++c) {           // classes 0..79, fully in range
        int k0 = 4 * c + 2 * h;
        v2f a;
        a[0] = vmask * __expf(lrow[k0]);
        a[1] = vmask * __expf(lrow[k0 + 1]);
        acc = __builtin_amdgcn_wmma_f32_16x16x4_f32(false, a, false, bones,
                                                    (short)0, acc, false, false);
    }
    {   // peeled chunk 20: only (h==0, class 80) is real; masked by multiply
        float m = (h == 0) ? vmask : 0.0f;
        v2f a;
        a[0] = m * __expf(lrow[(h == 0) ? 80 : 0]);
        a[1] = 0.0f;
        acc = __builtin_amdgcn_wmma_f32_16x16x4_f32(false, a, false, bones,
                                                    (short)0, acc, false, false);
    }
    // D layout (ISA 7.12.2): lanes 0-15 VGPR j -> M=j; lanes 16-31 VGPR j -> M=j+8.
    float cand  = vsel8(acc, lane & 7);
    float other = __shfl_xor(cand, 16, 32);
    bool  local = (lane < 8) || (lane >= 24);
    float denom = local ? cand : other;

    // ---- per-prior scalar work on half-wave h==0 (cold path, divergence OK) ----
    float loc_c = 0.0f, cep_c = 0.0f, pos_c = 0.0f;
    bool doit = (h == 0) && valid;
    if (doit) {
        float ovl = overlap_best[n * NP + p];
        int   obj = obj_best[n * NP + p];
        int   lab = labels[n * NO + obj];
        int   cls = (ovl < THRESH) ? 0 : lab;
        bool positive = (cls != 0);

        float b0 = boxes[(n * NO + obj) * 4 + 0], b1 = boxes[(n * NO + obj) * 4 + 1];
        float b2 = boxes[(n * NO + obj) * 4 + 2], b3 = boxes[(n * NO + obj) * 4 + 3];
        float bcx = 0.5f * (b0 + b2), bcy = 0.5f * (b1 + b3);
        float bw = b2 - b0, bh = b3 - b1;
        float pcx = priors_cxcy[p * 4 + 0], pcy = priors_cxcy[p * 4 + 1];
        float pw  = priors_cxcy[p * 4 + 2], ph  = priors_cxcy[p * 4 + 3];
        float g0 = (bcx - pcx) / (pw * 0.1f);
        float g1 = (bcy - pcy) / (ph * 0.1f);
        float g2 = __logf(bw / pw) * 5.0f;
        float g3 = __logf(bh / ph) * 5.0f;
        const float* lp = plocs + ((size_t)n * NP + p) * 4;
        float l1 = fabsf(lp[0] - g0) + fabsf(lp[1] - g1) +
                   fabsf(lp[2] - g2) + fabsf(lp[3] - g3);

        float cev = __logf(denom) - lrow[cls];   // CE from LDS-resident logits

        loc_c = positive ? l1 : 0.0f;
        cep_c = positive ? cev : 0.0f;
        pos_c = positive ? 1.0f : 0.0f;
        ce_neg[n * NP + p] = positive ? 0.0f : cev;
    }
    for (int off = 16; off >= 1; off >>= 1) {
        loc_c += __shfl_xor(loc_c, off, 32);
        cep_c += __shfl_xor(cep_c, off, 32);
        pos_c += __shfl_xor(pos_c, off, 32);
    }
    if (lane == 0) {
        if (loc_c != 0.0f) atomicAdd(&accs[0], loc_c);
        if (cep_c != 0.0f) atomicAdd(&accs[1], cep_c);
        if (pos_c != 0.0f) atomicAdd(&npos_row[n], pos_c);
    }
}

// ---------------- kernel 5: per-row descending bitonic sort + top-K sum -------------
__global__ void k_hardneg(const float* __restrict__ ce_neg,
                          const float* __restrict__ npos_row,
                          float* __restrict__ accs) {              // [2]=hard_neg
    extern __shared__ float smem[];
    __shared__ float red[256];
    const int n = blockIdx.x, t = threadIdx.x;
    for (int i = t; i < SORT_SZ; i += 256)
        smem[i] = (i < NP) ? ce_neg[n * NP + i] : -1.0f;   // CE >= 0, pad sinks
    __syncthreads();
    for (int k = 2; k <= SORT_SZ; k <<= 1) {
        for (int j = k >> 1; j > 0; j >>= 1) {
            for (int i = t; i < SORT_SZ; i += 256) {
                int ixj = i ^ j;
                if (ixj > i) {
                    bool down = ((i & k) == 0);
                    float a = smem[i], b = smem[ixj];
                    if (down ? (a < b) : (a > b)) { smem[i] = b; smem[ixj] = a; }
                }
            }
            __syncthreads();
        }
    }
    int K = 3 * (int)(npos_row[n] + 0.5f);
    if (K > NP) K = NP;
    float s = 0.0f;
    for (int i = t; i < K; i += 256) s += smem[i];
    red[t] = s;
    __syncthreads();
    for (int st = 128; st > 0; st >>= 1) {
        if (t < st) red[t] += red[t + st];
        __syncthreads();
    }
    if (t == 0) atomicAdd(&accs[2], red[0]);
}

// ---------------- kernel 6: finalize ----------------
__global__ void k_final(const float* __restrict__ accs,
                        const float* __restrict__ npos_row,
                        float* __restrict__ out) {
    float npt = 0.0f;
    for (int n = 0; n < NB; ++n) npt += npos_row[n];
    float conf = (accs[1] + accs[2]) / npt;
    float loc  = accs[0] / (4.0f * npt);
    out[0] = conf + loc;
}

extern "C" void kernel_launch(void* const* d_in, const int* in_sizes, int n_in,
                              void* d_out, int out_size, void* d_ws, size_t ws_size,
                              hipStream_t stream) {
    const float* plocs  = (const float*)d_in[0];   // (N,P,4)
    const float* scores = (const float*)d_in[1];   // (N,P,C)
    const float* boxes  = (const float*)d_in[2];   // (N,O,4)
    const int*   labels = (const int*)d_in[3];     // (N,O)
    const float* priors = (const float*)d_in[4];   // (P,4)
    float* out = (float*)d_out;

    char* w = (char*)d_ws;
    float* overlap_best = (float*)w;  w += (size_t)NB * NP * sizeof(float);
    int*   obj_best     = (int*)w;    w += (size_t)NB * NP * sizeof(int);
    float* ce_neg       = (float*)w;  w += (size_t)NB * NP * sizeof(float);
    int*   pfo          = (int*)w;    w += (size_t)NB * NO * sizeof(int);
    float* npos_row     = (float*)w;  w += (size_t)NB * sizeof(float);
    float* accs         = (float*)w;  // 4 floats

    k_init<<<1, 64, 0, stream>>>(npos_row, accs);
    k_overlap<<<dim3((NP + 255) / 256, NB), 256, 0, stream>>>(boxes, priors,
                                                              overlap_best, obj_best);
    k_pfo<<<dim3(NO, NB), 256, 0, stream>>>(boxes, priors, pfo);
    k_assign<<<1, NB, 0, stream>>>(pfo, overlap_best, obj_best);
    k_main<<<dim3((NP + PPB - 1) / PPB, NB), TPB, 0, stream>>>(
        plocs, scores, boxes, labels, priors, overlap_best, obj_best,
        ce_neg, npos_row, accs);
    k_hardneg<<<NB, 256, SORT_SZ * sizeof(float), stream>>>(ce_neg, npos_row, accs);
    k_final<<<1, 1, 0, stream>>>(accs, npos_row, out);
}